// OrthogonalLayer1D_66133906424510
// MI455X (gfx1250) — compile-verified
//
#include <hip/hip_runtime.h>

typedef __attribute__((ext_vector_type(2))) float    v2f;
typedef __attribute__((ext_vector_type(8))) float    v8f;
typedef __attribute__((ext_vector_type(4))) uint32_t v4u;
typedef __attribute__((ext_vector_type(8))) uint32_t v8u;

#define N_MODELS 16
#define DIM      256
#define N_POINTS 32768

__device__ __forceinline__ void lds_fence() {
  // CDNA5 split dep counter: order DS ops within the wave
  asm volatile("s_wait_dscnt 0" ::: "memory");
}

__device__ __forceinline__ uint32_t lds_addr(const void* p) {
  // generic pointers to LDS carry the LDS byte offset in the low 32 bits
  return (uint32_t)(uintptr_t)p;
}

// ---- Tensor DMA descriptor (ISA cdna5 §8.3/8.4), 2D tile: 16 rows x 256 f32,
// ---- row stride = N_POINTS*DIM elements. Groups 2/3 unused (<=2D tensor).
__device__ __forceinline__ v4u tdm_group0(uint32_t lds, uint64_t ga) {
  v4u g;
  g[0] = 1u;                                              // count=1: valid user D#
  g[1] = lds;                                             // lds_addr (bytes)
  g[2] = (uint32_t)ga;                                    // global_addr[31:0]
  g[3] = ((uint32_t)(ga >> 32) & 0x01FFFFFFu)             // global_addr[56:32]
         | (2u << 30);                                    // type = 2 ("image")
  return g;
}

__device__ __forceinline__ v8u tdm_group1() {
  v8u g;
  g[0] = 2u << 16;                      // workgroup_mask=0, data_size=2 (4 bytes)
  g[1] = (uint32_t)(DIM & 0xFFFF) << 16;       // tensor_dim0[15:0] = 256
  g[2] = ((uint32_t)DIM >> 16)                 // tensor_dim0[31:16] = 0
         | ((uint32_t)N_MODELS << 16);         // tensor_dim1[15:0]  = 16
  g[3] = (uint32_t)DIM << 16;                  // tensor_dim1 hi=0, tile_dim0 = 256
  g[4] = (uint32_t)N_MODELS;                   // tile_dim1 = 16, tile_dim2 = 0
  g[5] = (uint32_t)(N_POINTS * DIM);           // tensor_dim0_stride[31:0] = 2^23
  g[6] = 0u;                                   // stride hi, tensor_dim1_stride lo
  g[7] = 0u;                                   // tensor_dim1_stride hi
  return g;
}

__global__ __launch_bounds__(256, 4)
void OrthogonalLayer1D_gs_wmma(const float* __restrict__ x,
                               float* __restrict__ out,
                               int n_points) {
  // One workgroup (8 waves) per point.
  __shared__ __align__(16) float sV[N_MODELS * DIM]; // 16 KB: input tile V
  __shared__ __align__(16) float sO[N_MODELS * DIM]; // 16 KB: output staging
  __shared__ float sG[16 * 16];                      // Gram -> Cholesky L
  __shared__ float sW[16 * 16];                      // W = L^-1

  const int tid  = threadIdx.x;
  const int wave = tid >> 5;
  const int lane = tid & 31;
  const int p    = blockIdx.x;
  if (p >= n_points) return; // uniform per block

  // --- zero the Gram accumulator (16*16 == 256 == blockDim) ---
  sG[tid] = 0.0f;

  // --- stage V (16x256 f32, row stride 2^23 elems) via one TDM descriptor ---
  if (wave == 0) {
    const uint64_t ga = (uint64_t)(uintptr_t)(x + (size_t)p * DIM);
    const uint32_t l  = (uint32_t)__builtin_amdgcn_readfirstlane(lds_addr(sV));
    const v4u g0 = tdm_group0(l, ga);
    const v8u g1 = tdm_group1();
    asm volatile("tensor_load_to_lds %0, %1" :: "s"(g0), "s"(g1) : "memory");
    __builtin_amdgcn_s_wait_tensorcnt(0);
  }
  __syncthreads();

  // --- G += V V^T via V_WMMA_F32_16X16X4_F32; K=256 split over 8 waves ---
  // A 16x4 f32 layout: lanes 0-15 hold (K=k0,k0+1), lanes 16-31 hold (K=k0+2,k0+3)
  const int row  = lane & 15;
  const int koff = (lane >> 4) << 1;
  v8f g = {};
#pragma unroll
  for (int t = 0; t < 8; ++t) {
    const int k0 = (wave * 8 + t) * 4;
    v2f a;
    a.x = sV[row * DIM + k0 + koff];
    a.y = sV[row * DIM + k0 + koff + 1];
    // symmetric product: B register tile == A register tile
    g = __builtin_amdgcn_wmma_f32_16x16x4_f32(false, a, false, a, (short)0, g,
                                              false, false);
  }
  // reduce partial Gram matrices across waves with ds_add_f32 atomics
  {
    const int half = lane >> 4;
#pragma unroll
    for (int r = 0; r < 8; ++r) {
      const int mrow = r + (half << 3); // C/D layout: VGPR r -> M=r or M=r+8
      atomicAdd(&sG[mrow * 16 + (lane & 15)], g[r]);
    }
  }
  __syncthreads();

  // --- wave 0: Cholesky G = L L^T (lower, in place), then W = L^-1 ---
  if (wave == 0) {
    const int i = lane;
    for (int k = 0; k < 16; ++k) {
      const float dkk = sG[k * 16 + k];
      const float dk  = sqrtf(fmaxf(dkk, 0.0f));
      const float inv = dk > 0.0f ? 1.0f / dk : 0.0f; // safe_div: zero column
      if (i == k) sG[k * 16 + k] = dk;
      if (i < 16 && i > k) sG[i * 16 + k] *= inv;
      lds_fence();
      if (i < 16 && i > k) {
        const float lik = sG[i * 16 + k];
        for (int j = k + 1; j <= i; ++j)
          sG[i * 16 + j] -= lik * sG[j * 16 + k];
      }
      lds_fence();
    }
    // lane j computes column j of W = L^-1 (forward substitution)
    if (i < 16) {
      const int j = i;
      const float djj = sG[j * 16 + j];
      for (int r = 0; r < j; ++r) sW[r * 16 + j] = 0.0f;
      sW[j * 16 + j] = djj > 0.0f ? 1.0f / djj : 0.0f;
      lds_fence();
      for (int r = j + 1; r < 16; ++r) {
        float s = 0.0f;
        for (int k = j; k < r; ++k) s += sG[r * 16 + k] * sW[k * 16 + j];
        const float drr = sG[r * 16 + r];
        sW[r * 16 + j] = drr > 0.0f ? -s / drr : 0.0f;
        lds_fence();
      }
    }
  }
  __syncthreads();

  // --- B = W * V : each wave emits two 16-column tiles via 4 chained WMMAs ---
#pragma unroll
  for (int tt = 0; tt < 2; ++tt) {
    const int n0 = (wave * 2 + tt) * 16;
    v8f c = {};
#pragma unroll
    for (int ks = 0; ks < 4; ++ks) {
      const int k0 = ks * 4;
      v2f a, b;
      // A tile: W[:, k0:k0+4] in 16x4 layout
      a.x = sW[row * 16 + k0 + koff];
      a.y = sW[row * 16 + k0 + koff + 1];
      // B tile 4x16: lanes 0-15 hold rows k0,k0+1; lanes 16-31 hold k0+2,k0+3
      b.x = sV[(k0 + koff) * DIM + n0 + row];
      b.y = sV[(k0 + koff + 1) * DIM + n0 + row];
      c = __builtin_amdgcn_wmma_f32_16x16x4_f32(false, a, false, b, (short)0, c,
                                                false, false);
    }
    // stage C/D layout into LDS output tile (row-major sO[m][n])
    const int n    = n0 + row;
    const int half = lane >> 4;
#pragma unroll
    for (int r = 0; r < 8; ++r) {
      const int m = r + (half << 3);
      sO[m * DIM + n] = c[r];
    }
  }
  __syncthreads(); // all sO writes visible before the TDM reads LDS

  // --- stream the 16 KB output tile out with one TDM store descriptor ---
  if (wave == 0) {
    const uint64_t ga = (uint64_t)(uintptr_t)(out + (size_t)p * DIM);
    const uint32_t l  = (uint32_t)__builtin_amdgcn_readfirstlane(lds_addr(sO));
    const v4u g0 = tdm_group0(l, ga);
    const v8u g1 = tdm_group1();
    asm volatile("tensor_store_from_lds %0, %1" :: "s"(g0), "s"(g1) : "memory");
    __builtin_amdgcn_s_wait_tensorcnt(0); // retire before wave end
  }
}

extern "C" void kernel_launch(void* const* d_in, const int* in_sizes, int n_in,
                              void* d_out, int out_size, void* d_ws, size_t ws_size,
                              hipStream_t stream) {
  (void)in_sizes; (void)n_in; (void)out_size; (void)d_ws; (void)ws_size;
  const float* x = (const float*)d_in[0];
  float* out = (float*)d_out;
  OrthogonalLayer1D_gs_wmma<<<N_POINTS, 256, 0, stream>>>(x, out, N_POINTS);
}